// GANN_crys_fea_67577015435345
// MI455X (gfx1250) — compile-verified
//
#include <hip/hip_runtime.h>
#include <hip/hip_bf16.h>
#include <math.h>

#define N_ATOMS 1048576
#define N_CRYS  8192
#define NFEA    128
#define NCLS    5
#define LN_EPS  1e-5f

typedef float v2f __attribute__((ext_vector_type(2)));
typedef float v4f __attribute__((ext_vector_type(4)));
typedef float v8f __attribute__((ext_vector_type(8)));

__device__ __forceinline__ float softplus_f(float x) {
    // logaddexp(x, 0) = max(x,0) + log1p(exp(-|x|))  (matches jax.nn.softplus)
    return fmaxf(x, 0.0f) + log1pf(expf(-fabsf(x)));
}

// ---------------------------------------------------------------------------
// Kernel 1: segment-mean pool + LayerNorm + softplus.
// One wave32 per crystal. Lane l owns features [4l, 4l+3] (one float4).
// Each atom row = one fully coalesced 512B wavefront read (b128 per lane).
// Non-temporal loads: 512 MB streamed once, > L2 (192 MB), don't cache it.
// ---------------------------------------------------------------------------
__global__ __launch_bounds__(256) void pool_ln_softplus_kernel(
    const float* __restrict__ fea,   // (N_ATOMS, 128)
    const int*   __restrict__ idx,   // (N_CRYS, 2) [start, end)
    const float* __restrict__ ln_w,  // (128)
    const float* __restrict__ ln_b,  // (128)
    float*       __restrict__ act)   // (N_CRYS, 128) output
{
    const int wave = (int)((blockIdx.x * blockDim.x + threadIdx.x) >> 5);
    const int lane = (int)(threadIdx.x & 31);
    if (wave >= N_CRYS) return;

    const int start = idx[2 * wave + 0];
    const int end   = idx[2 * wave + 1];
    const float inv_len = 1.0f / (float)(end - start);

    const v4f* __restrict__ p = (const v4f*)fea + (size_t)start * 32 + lane;

    v4f acc0 = {0.f, 0.f, 0.f, 0.f};
    v4f acc1 = acc0, acc2 = acc0, acc3 = acc0;

    int r = start;
    // 4-deep unroll: keep 4 global_load_b128 in flight per wave.
    for (; r + 4 <= end; r += 4, p += 128) {
        acc0 += __builtin_nontemporal_load(p);
        acc1 += __builtin_nontemporal_load(p + 32);
        acc2 += __builtin_nontemporal_load(p + 64);
        acc3 += __builtin_nontemporal_load(p + 96);
    }
    for (; r < end; ++r, p += 32) {
        acc0 += __builtin_nontemporal_load(p);
    }
    v4f pooled = ((acc0 + acc1) + (acc2 + acc3)) * inv_len;

    // Wave-wide mean / variance over the 128 features.
    float s  = pooled.x + pooled.y + pooled.z + pooled.w;
    float s2 = pooled.x * pooled.x + pooled.y * pooled.y +
               pooled.z * pooled.z + pooled.w * pooled.w;
    #pragma unroll
    for (int m = 16; m >= 1; m >>= 1) {
        s  += __shfl_xor(s,  m, 32);
        s2 += __shfl_xor(s2, m, 32);
    }
    const float mu   = s * (1.0f / (float)NFEA);
    const float var  = s2 * (1.0f / (float)NFEA) - mu * mu;
    const float rstd = rsqrtf(var + LN_EPS);

    const v4f w = *((const v4f*)ln_w + lane);
    const v4f b = *((const v4f*)ln_b + lane);

    v4f y;
    y.x = softplus_f((pooled.x - mu) * rstd * w.x + b.x);
    y.y = softplus_f((pooled.y - mu) * rstd * w.y + b.y);
    y.z = softplus_f((pooled.z - mu) * rstd * w.z + b.z);
    y.w = softplus_f((pooled.w - mu) * rstd * w.w + b.w);

    *((v4f*)act + (size_t)wave * 32 + lane) = y;   // coalesced 512B store
}

// ---------------------------------------------------------------------------
// Kernel 2: my_output = act @ out_w^T + out_b via V_WMMA_F32_16X16X4_F32.
// N padded 5 -> 16 by staging out_w zero-padded in LDS (8 KB), so every
// B-fragment fetch is an unconditional ds_load_b64 (no EXEC predication).
// One wave per 16-row tile (512 tiles), K=128 as a chain of 32 f32 WMMAs.
// ---------------------------------------------------------------------------
__global__ __launch_bounds__(256) void head_wmma_kernel(
    const float* __restrict__ act,    // (N_CRYS, 128)
    const float* __restrict__ out_w,  // (5, 128)
    const float* __restrict__ out_b,  // (5)
    float*       __restrict__ out)    // (N_CRYS, 5)
{
    __shared__ float wsh[16 * NFEA];  // out_w padded to 16 rows, rows 5..15 = 0

    const int tid = (int)threadIdx.x;
    #pragma unroll
    for (int i = tid; i < 16 * NFEA; i += 256) {
        const int n = i >> 7;          // row (class), 0..15
        wsh[i] = (n < NCLS) ? out_w[i] : 0.0f;
    }
    __syncthreads();

    const int wave = (int)((blockIdx.x * blockDim.x + threadIdx.x) >> 5);
    const int lane = tid & 31;
    const int tile = wave;                 // 0 .. 511
    if (tile >= N_CRYS / 16) return;

    const int half = lane >> 4;            // 0: lanes 0-15, 1: lanes 16-31
    const int l16  = lane & 15;

    // A row owned by this lane (ISA A-layout, 16x4 f32):
    //   lanes 0-15 hold (M=lane, K={0,1}); lanes 16-31 hold (M=lane-16, K={2,3}).
    const float* __restrict__ arow = act + (size_t)(tile * 16 + l16) * NFEA;
    // B column n = l16 from the zero-padded LDS copy (unconditional loads).
    const float* __restrict__ wrow = &wsh[l16 * NFEA];

    v8f c = {0.f, 0.f, 0.f, 0.f, 0.f, 0.f, 0.f, 0.f};

    #pragma unroll
    for (int kk = 0; kk < NFEA / 4; ++kk) {
        const int k0 = kk * 4 + half * 2;  // per-half K offsets {0,1} / {2,3}
        v2f a = *(const v2f*)(arow + k0);  // global_load_b64
        v2f b = *(const v2f*)(wrow + k0);  // ds_load_b64
        c = __builtin_amdgcn_wmma_f32_16x16x4_f32(
                /*neg_a=*/false, a, /*neg_b=*/false, b,
                /*c_mod=*/(short)0, c, /*reuse_a=*/false, /*reuse_b=*/false);
    }

    const bool  valid = (l16 < NCLS);
    const float bias  = valid ? out_b[l16] : 0.0f;

    // C/D layout: VGPR r -> (M = r + 8*half, N = l16).
    #pragma unroll
    for (int rr = 0; rr < 8; ++rr) {
        const int row = tile * 16 + rr + half * 8;
        if (valid) out[(size_t)row * NCLS + l16] = c[rr] + bias;
    }
}

extern "C" void kernel_launch(void* const* d_in, const int* in_sizes, int n_in,
                              void* d_out, int out_size, void* d_ws, size_t ws_size,
                              hipStream_t stream) {
    (void)in_sizes; (void)n_in; (void)d_ws; (void)ws_size; (void)out_size;

    const float* fea   = (const float*)d_in[0];  // (N_ATOMS, 128)
    const int*   idx   = (const int*)  d_in[1];  // (N_CRYS, 2)
    const float* ln_w  = (const float*)d_in[2];  // (128)
    const float* ln_b  = (const float*)d_in[3];  // (128)
    const float* out_w = (const float*)d_in[4];  // (5, 128)
    const float* out_b = (const float*)d_in[5];  // (5)

    float* my_output = (float*)d_out;                         // (N_CRYS, 5)
    float* act       = (float*)d_out + (size_t)N_CRYS * NCLS; // (N_CRYS, 128)

    // Kernel 1: one wave per crystal; 256 threads = 8 waves per block.
    {
        const int waves_per_block = 256 / 32;
        const int blocks = N_CRYS / waves_per_block;   // 1024
        pool_ln_softplus_kernel<<<blocks, 256, 0, stream>>>(fea, idx, ln_w, ln_b, act);
    }
    // Kernel 2: one wave per 16-row tile; 512 tiles -> 64 blocks of 256.
    {
        const int tiles = N_CRYS / 16;                 // 512
        const int waves_per_block = 256 / 32;
        const int blocks = tiles / waves_per_block;    // 64
        head_wmma_kernel<<<blocks, 256, 0, stream>>>(act, out_w, out_b, my_output);
    }
}